// cls_model_75874892251826
// MI455X (gfx1250) — compile-verified
//
#include <hip/hip_runtime.h>

typedef __attribute__((ext_vector_type(2))) float v2f;
typedef __attribute__((ext_vector_type(8))) float v8f;

#define BB   16      // batch
#define NN   2048    // points
#define FIN  6       // in features
#define FPAD 8       // padded in features (for K=4 WMMA steps)
#define FOUT 128     // out features
#define KCH  6       // Chebyshev order
#define KNN  40      // top-k
#define NCLS 40      // classes
#define NEG_INF (-1e30f)

// ---------------------------------------------------------------------------
// 0) Pad x -> [N][8] (zeros in cols 6,7) and precompute |x|^2 per node.
//    Makes all WMMA operand loads contiguous b64, no divergent predication.
// ---------------------------------------------------------------------------
__global__ void padx_kernel(const float* __restrict__ x, float* __restrict__ xpad,
                            float* __restrict__ sqb) {
  const int i = blockIdx.x * 256 + threadIdx.x;   // over B*N nodes
  if (i >= BB * NN) return;
  float s = 0.f;
#pragma unroll
  for (int f = 0; f < FIN; ++f) {
    const float v = x[(size_t)i * FIN + f];
    xpad[(size_t)i * FPAD + f] = v;
    s += v * v;
  }
  xpad[(size_t)i * FPAD + 6] = 0.f;
  xpad[(size_t)i * FPAD + 7] = 0.f;
  sqb[i] = s;
}

// ---------------------------------------------------------------------------
// 1) Pairwise adjacency tile via V_WMMA_F32_16X16X4_F32.
//    adj[i,j] = exp(-(|xi|^2 - 2 xi.xj + |xj|^2)). K padded to 8 (2 WMMAs).
//    One wave per 16x16 tile. A-layout (16x4 f32): lane l holds M=l&15,
//    VGPR0/1 = K = 2*(l>>4) + {0,1}. B-layout (4x16) mirrored.
// ---------------------------------------------------------------------------
__global__ __launch_bounds__(32)
void adj_wmma_kernel(const float* __restrict__ xpad, const float* __restrict__ sqb,
                     float* __restrict__ adj) {
  const int ti = blockIdx.x << 4;   // row tile base
  const int tj = blockIdx.y << 4;   // col tile base
  const int l  = threadIdx.x;
  const int lh = l >> 4;            // half-wave id (0/1)
  const int ll = l & 15;

  __shared__ float sqi[16], sqj[16];
  if (l < 16) { sqi[l] = sqb[ti + l]; sqj[l] = sqb[tj + l]; }
  __syncthreads();

  v8f c = {};
#pragma unroll
  for (int ks = 0; ks < 2; ++ks) {
    const int k0 = ks * 4 + lh * 2;                               // 8B aligned
    const v2f a = *(const v2f*)(xpad + (size_t)(ti + ll) * FPAD + k0);
    const v2f b = *(const v2f*)(xpad + (size_t)(tj + ll) * FPAD + k0);
    c = __builtin_amdgcn_wmma_f32_16x16x4_f32(false, a, false, b,
                                              (short)0, c, false, false);
  }
  // D element (M = v + 8*lh, N = ll)
#pragma unroll
  for (int v = 0; v < 8; ++v) {
    const int m = v + 8 * lh;
    adj[(size_t)(ti + m) * NN + (tj + ll)] =
        __expf(-(sqi[m] - 2.f * c[v] + sqj[ll]));
  }
}

// ---------------------------------------------------------------------------
// 2) Row-wise top-40 selection + row-sum d. One block (256 thr) per row.
// ---------------------------------------------------------------------------
__global__ __launch_bounds__(256)
void topk_kernel(const float* __restrict__ adj, float* __restrict__ topv,
                 int* __restrict__ topi, float* __restrict__ dsum) {
  const int row = blockIdx.x;
  const int t   = threadIdx.x;
  __shared__ float rv[NN];
  __shared__ float redv[256];
  __shared__ int   redi[256];
  __shared__ float dacc;

  const float* ar = adj + (size_t)row * NN;
  for (int j = t; j < NN; j += 256) rv[j] = ar[j];
  if (t == 0) dacc = 0.f;
  __syncthreads();

  for (int it = 0; it < KNN; ++it) {
    float bm = NEG_INF; int bi = 0;
#pragma unroll
    for (int u = 0; u < NN / 256; ++u) {
      const int j = t + u * 256;
      const float v = rv[j];
      if (v > bm) { bm = v; bi = j; }
    }
    redv[t] = bm; redi[t] = bi;
    __syncthreads();
    for (int s = 128; s > 0; s >>= 1) {
      if (t < s && redv[t + s] > redv[t]) { redv[t] = redv[t + s]; redi[t] = redi[t + s]; }
      __syncthreads();
    }
    if (t == 0) {
      const float v = redv[0]; const int j = redi[0];
      topv[(size_t)row * KNN + it] = v;
      topi[(size_t)row * KNN + it] = j;
      dacc += v;
      rv[j] = NEG_INF;
    }
    __syncthreads();
  }
  if (t == 0) dsum[row] = dacc;
}

// ---------------------------------------------------------------------------
// 3) dinv = d>0 ? 1/sqrt(d) : 0   over all B*N rows
// ---------------------------------------------------------------------------
__global__ void dinv_kernel(const float* __restrict__ d, float* __restrict__ dinv) {
  const int i = blockIdx.x * 256 + threadIdx.x;
  if (i >= BB * NN) return;
  const float v = d[i];
  dinv[i] = (v > 0.f) ? rsqrtf(v) : 0.f;
}

// x -> xk[b][0]
__global__ void copyx0_kernel(const float* __restrict__ x, float* __restrict__ xk) {
  const int i = blockIdx.x * 256 + threadIdx.x;
  if (i >= BB * NN * FIN) return;
  const int b = i / (NN * FIN), r = i % (NN * FIN);
  xk[((size_t)b * KCH) * NN * FIN + r] = x[i];
}

// ---------------------------------------------------------------------------
// 4) Chebyshev step on [N,6] (sparse L = I - D^-1/2 A D^-1/2):
//    xout = alpha*(xin - dinv_n * sum_t w_t*dinv_j*xin[j]) + beta*xprev
// ---------------------------------------------------------------------------
__global__ __launch_bounds__(256)
void cheb_spmv_kernel(const float* __restrict__ xin, const float* __restrict__ xprev,
                      float* __restrict__ xout, const float* __restrict__ topv,
                      const int* __restrict__ topi, const float* __restrict__ dinv,
                      float alpha, float beta) {
  const int n = blockIdx.x * 256 + threadIdx.x;
  if (n >= NN) return;
  float acc[FIN] = {0.f, 0.f, 0.f, 0.f, 0.f, 0.f};
  const float di = dinv[n];
  for (int t = 0; t < KNN; ++t) {
    const float w = topv[(size_t)n * KNN + t];
    const int   j = topi[(size_t)n * KNN + t];
    const float wj = w * dinv[j];
#pragma unroll
    for (int f = 0; f < FIN; ++f) acc[f] += wj * xin[(size_t)j * FIN + f];
  }
#pragma unroll
  for (int f = 0; f < FIN; ++f) {
    const float y = xin[(size_t)n * FIN + f] - di * acc[f];
    xout[(size_t)n * FIN + f] = alpha * y + beta * xprev[(size_t)n * FIN + f];
  }
}

// ---------------------------------------------------------------------------
// 5) out[b,n,o] = relu(b1[o] + sum_{k,f} xk[b,k,n,f] * W1[k,f,o])
// ---------------------------------------------------------------------------
__global__ __launch_bounds__(FOUT)
void chebout_kernel(const float* __restrict__ xk, const float* __restrict__ W1,
                    const float* __restrict__ b1, float* __restrict__ outf) {
  const int bn = blockIdx.x;
  const int b = bn >> 11, n = bn & (NN - 1);
  const int o = threadIdx.x;
  __shared__ float xl[KCH * FIN];
  if (o < KCH * FIN) {
    const int kk = o / FIN, f = o % FIN;
    xl[o] = xk[(((size_t)b * KCH + kk) * NN + n) * FIN + f];
  }
  __syncthreads();
  float acc = b1[o];
#pragma unroll
  for (int kf = 0; kf < KCH * FIN; ++kf) acc += xl[kf] * W1[(size_t)kf * FOUT + o];
  outf[((size_t)b * NN + n) * FOUT + o] = fmaxf(acc, 0.f);
}

// ---------------------------------------------------------------------------
// 6) Lout = L @ out (sparse gather over 40 neighbors, 128 feats per node)
// ---------------------------------------------------------------------------
__global__ __launch_bounds__(FOUT)
void spmv128_kernel(const float* __restrict__ outf, const float* __restrict__ topv,
                    const int* __restrict__ topi, const float* __restrict__ dinv,
                    float* __restrict__ loutf) {
  const int bn = blockIdx.x;
  const int b = bn >> 11, n = bn & (NN - 1);
  const int o = threadIdx.x;
  __shared__ float wsh[KNN];
  __shared__ int   jsh[KNN];
  const float di = dinv[bn];
  if (o < KNN) {
    const int j = topi[(size_t)bn * KNN + o];
    wsh[o] = di * topv[(size_t)bn * KNN + o] * dinv[(b << 11) + j];
    jsh[o] = j;
  }
  __syncthreads();
  const float* ob = outf + (size_t)b * NN * FOUT;
  float acc = 0.f;
  for (int t = 0; t < KNN; ++t) acc += wsh[t] * ob[(size_t)jsh[t] * FOUT + o];
  loutf[(size_t)bn * FOUT + o] = ob[(size_t)n * FOUT + o] - acc;
}

// ---------------------------------------------------------------------------
// 6b) Transpose [B][NN][FOUT] -> [B][FOUT][NN] (LDS tiled, conflict-free)
// ---------------------------------------------------------------------------
__global__ __launch_bounds__(256)
void transpose_kernel(const float* __restrict__ src, float* __restrict__ dst) {
  __shared__ float tile[32][33];
  const int b  = blockIdx.z;
  const int n0 = blockIdx.x * 32;     // along NN
  const int f0 = blockIdx.y * 32;     // along FOUT
  const int tx = threadIdx.x & 31, ty = threadIdx.x >> 5;   // 32 x 8
  const float* s = src + (size_t)b * NN * FOUT;
  float* d       = dst + (size_t)b * FOUT * NN;
#pragma unroll
  for (int r = 0; r < 32; r += 8)
    tile[ty + r][tx] = s[(size_t)(n0 + ty + r) * FOUT + f0 + tx];
  __syncthreads();
#pragma unroll
  for (int r = 0; r < 32; r += 8)
    d[(size_t)(f0 + ty + r) * NN + n0 + tx] = tile[tx][ty + r];
}

// ---------------------------------------------------------------------------
// 7) M[b] = out[b]^T @ Lout[b] via f32 WMMA on transposed ([FOUT][NN]) data.
//    4 waves/block, one 16x16 tile per wave, contiguous b64 operand loads,
//    periodic global_prefetch on the K-stream.
// ---------------------------------------------------------------------------
__global__ __launch_bounds__(128)
void mgemm_wmma_kernel(const float* __restrict__ oT, const float* __restrict__ lT,
                       float* __restrict__ M) {
  const int b  = blockIdx.z;
  const int f0 = blockIdx.x << 4;                                   // 8 tiles
  const int g0 = ((blockIdx.y << 2) + (threadIdx.x >> 5)) << 4;     // 2*4 tiles
  const int l  = threadIdx.x & 31, lh = l >> 4, ll = l & 15;

  const float* oa = oT + (size_t)b * FOUT * NN + (size_t)(f0 + ll) * NN + lh * 2;
  const float* ba = lT + (size_t)b * FOUT * NN + (size_t)(g0 + ll) * NN + lh * 2;

  v8f c = {};
  for (int kb = 0; kb < NN; kb += 8) {
    if ((kb & 127) == 0 && kb + 256 < NN) {
      __builtin_prefetch(oa + kb + 256, 0, 3);
      __builtin_prefetch(ba + kb + 256, 0, 3);
    }
#pragma unroll
    for (int s = 0; s < 2; ++s) {
      const v2f a  = *(const v2f*)(oa + kb + s * 4);
      const v2f bv = *(const v2f*)(ba + kb + s * 4);
      c = __builtin_amdgcn_wmma_f32_16x16x4_f32(false, a, false, bv,
                                                (short)0, c, false, false);
    }
  }
  float* Mb = M + (size_t)b * FOUT * FOUT;
#pragma unroll
  for (int v = 0; v < 8; ++v)
    Mb[(size_t)(f0 + v + 8 * lh) * FOUT + g0 + ll] = c[v];
}

// ---------------------------------------------------------------------------
// 8) reg = sum(M*M): deterministic two-stage reduction (no atomics)
// ---------------------------------------------------------------------------
__global__ __launch_bounds__(256)
void regpart_kernel(const float* __restrict__ M, float* __restrict__ part) {
  __shared__ float red[256];
  const float v = M[(size_t)blockIdx.x * 256 + threadIdx.x];
  red[threadIdx.x] = v * v;
  __syncthreads();
  for (int s = 128; s > 0; s >>= 1) {
    if (threadIdx.x < s) red[threadIdx.x] += red[threadIdx.x + s];
    __syncthreads();
  }
  if (threadIdx.x == 0) part[blockIdx.x] = red[0];
}

__global__ __launch_bounds__(256)
void regfinal_kernel(const float* __restrict__ part, float* __restrict__ outp) {
  __shared__ float red[256];
  float s = 0.f;
  for (int i = threadIdx.x; i < 1024; i += 256) s += part[i];
  red[threadIdx.x] = s;
  __syncthreads();
  for (int st = 128; st > 0; st >>= 1) {
    if (threadIdx.x < st) red[threadIdx.x] += red[threadIdx.x + st];
    __syncthreads();
  }
  if (threadIdx.x == 0) outp[BB * NCLS] = red[0];
}

// ---------------------------------------------------------------------------
// 9) max-pool over N then FC: logits[b,c] = pooled . fc_w[:,c] + fc_b[c]
// ---------------------------------------------------------------------------
__global__ __launch_bounds__(FOUT)
void poolfc_kernel(const float* __restrict__ outf, const float* __restrict__ fcw,
                   const float* __restrict__ fcb, float* __restrict__ outp) {
  const int b = blockIdx.x, t = threadIdx.x;
  __shared__ float pl[FOUT];
  const float* ob = outf + (size_t)b * NN * FOUT;
  float m = NEG_INF;
  for (int n = 0; n < NN; ++n) m = fmaxf(m, ob[(size_t)n * FOUT + t]);
  pl[t] = m;
  __syncthreads();
  if (t < NCLS) {
    float acc = fcb[t];
#pragma unroll 8
    for (int o = 0; o < FOUT; ++o) acc += pl[o] * fcw[(size_t)o * NCLS + t];
    outp[b * NCLS + t] = acc;
  }
}

// ---------------------------------------------------------------------------
// Host orchestration
// ---------------------------------------------------------------------------
static inline size_t alignup(size_t v) { return (v + 255) & ~(size_t)255; }

extern "C" void kernel_launch(void* const* d_in, const int* in_sizes, int n_in,
                              void* d_out, int out_size, void* d_ws, size_t ws_size,
                              hipStream_t stream) {
  const float* x    = (const float*)d_in[0];   // [B,N,FIN]
  const float* W1   = (const float*)d_in[5];   // [K,FIN,FOUT]
  const float* b1   = (const float*)d_in[6];   // [FOUT]
  const float* fc_w = (const float*)d_in[7];   // [FOUT,NCLS]
  const float* fc_b = (const float*)d_in[8];   // [NCLS]
  float* outp = (float*)d_out;                 // 640 logits + 1 reg
  (void)in_sizes; (void)n_in; (void)out_size; (void)ws_size;

  char* w = (char*)d_ws;
  size_t off = 0;
  auto take = [&](size_t bytes) { void* p = w + off; off = alignup(off + bytes); return p; };
  float* adjbuf = (float*)take((size_t)NN * NN * 4);            // per-batch, reused
  float* xpad   = (float*)take((size_t)BB * NN * FPAD * 4);
  float* sqb    = (float*)take((size_t)BB * NN * 4);
  float* dsumb  = (float*)take((size_t)BB * NN * 4);
  float* dinvb  = (float*)take((size_t)BB * NN * 4);
  float* topv   = (float*)take((size_t)BB * NN * KNN * 4);
  int*   topi   = (int*)  take((size_t)BB * NN * KNN * 4);
  float* xk     = (float*)take((size_t)BB * KCH * NN * FIN * 4);
  float* outf   = (float*)take((size_t)BB * NN * FOUT * 4);
  float* loutf  = (float*)take((size_t)BB * NN * FOUT * 4);
  float* outT   = (float*)take((size_t)BB * NN * FOUT * 4);
  float* loutT  = (float*)take((size_t)BB * NN * FOUT * 4);
  float* Mbuf   = (float*)take((size_t)BB * FOUT * FOUT * 4);
  float* regp   = (float*)take(1024 * 4);

  // Pad features + |x|^2, then adjacency (WMMA) + top-k per batch (adjbuf reuse)
  padx_kernel<<<(BB * NN + 255) / 256, 256, 0, stream>>>(x, xpad, sqb);
  for (int b = 0; b < BB; ++b) {
    adj_wmma_kernel<<<dim3(NN / 16, NN / 16), 32, 0, stream>>>(
        xpad + (size_t)b * NN * FPAD, sqb + (size_t)b * NN, adjbuf);
    topk_kernel<<<NN, 256, 0, stream>>>(
        adjbuf, topv + (size_t)b * NN * KNN, topi + (size_t)b * NN * KNN,
        dsumb + (size_t)b * NN);
  }
  dinv_kernel<<<(BB * NN + 255) / 256, 256, 0, stream>>>(dsumb, dinvb);
  copyx0_kernel<<<(BB * NN * FIN + 255) / 256, 256, 0, stream>>>(x, xk);

  // Chebyshev recursion: x_k = 2 L x_{k-1} - x_{k-2} (x_1 = L x_0)
  for (int b = 0; b < BB; ++b) {
    float* xb = xk + (size_t)b * KCH * NN * FIN;
    const float* tvb = topv + (size_t)b * NN * KNN;
    const int*   tib = topi + (size_t)b * NN * KNN;
    const float* dvb = dinvb + (size_t)b * NN;
    cheb_spmv_kernel<<<NN / 256, 256, 0, stream>>>(
        xb, xb, xb + (size_t)NN * FIN, tvb, tib, dvb, 1.f, 0.f);
    for (int k = 2; k < KCH; ++k)
      cheb_spmv_kernel<<<NN / 256, 256, 0, stream>>>(
          xb + (size_t)(k - 1) * NN * FIN, xb + (size_t)(k - 2) * NN * FIN,
          xb + (size_t)k * NN * FIN, tvb, tib, dvb, 2.f, -1.f);
  }

  chebout_kernel<<<BB * NN, FOUT, 0, stream>>>(xk, W1, b1, outf);
  spmv128_kernel<<<BB * NN, FOUT, 0, stream>>>(outf, topv, topi, dinvb, loutf);

  transpose_kernel<<<dim3(NN / 32, FOUT / 32, BB), 256, 0, stream>>>(outf, outT);
  transpose_kernel<<<dim3(NN / 32, FOUT / 32, BB), 256, 0, stream>>>(loutf, loutT);
  mgemm_wmma_kernel<<<dim3(FOUT / 16, FOUT / 64, BB), 128, 0, stream>>>(outT, loutT, Mbuf);

  regpart_kernel<<<(BB * FOUT * FOUT) / 256, 256, 0, stream>>>(Mbuf, regp);
  regfinal_kernel<<<1, 256, 0, stream>>>(regp, outp);
  poolfc_kernel<<<BB, FOUT, 0, stream>>>(outf, fc_w, fc_b, outp);
}